// ViT_SAPS_19877108646374
// MI455X (gfx1250) — compile-verified
//
#include <hip/hip_runtime.h>
#include <stdint.h>

typedef __attribute__((ext_vector_type(2))) float    v2f;
typedef __attribute__((ext_vector_type(8))) float    v8f;
typedef __attribute__((ext_vector_type(4))) uint32_t v4u;
typedef __attribute__((ext_vector_type(4))) int      v4i;
typedef __attribute__((ext_vector_type(8))) int      v8i;

#define NBATCH 4
#define DMODEL 768
#define IMGSZ  1024
#define NCELLS 256                      // 4px cells per side
#define CELLS  (NCELLS*NCELLS)          // 65536 cells / image
#define PYRSTR 87424                    // pyramid floats per image (padded)
#define LISTPI 87360                    // list entries per image (worst case)

#if __has_builtin(__builtin_amdgcn_tensor_load_to_lds) && __has_builtin(__builtin_amdgcn_s_wait_tensorcnt)
#define USE_TDM 1
#else
#define USE_TDM 0
#endif

// pyramid level offsets (level k holds (256>>k)^2 sums of (4<<k)-sized blocks)
__device__ const int g_lvloff[9] = {0,65536,81920,86016,87040,87296,87360,87376,87380};
static const int h_lvloff[9]     = {0,65536,81920,86016,87040,87296,87360,87376,87380};
// per-size list capacity prefix (sizes 4,8,16,32,64,128)
__device__ const int g_capoff[6] = {0,65536,81920,86016,87040,87296};

// ---- workspace layout (bytes) ----
static constexpr size_t OFF_PYR   = 0;
static constexpr size_t OFF_LEAF  = OFF_PYR  + (size_t)NBATCH*PYRSTR*4;   // int[4*65536] packed size/cy/cx (Morton)
static constexpr size_t OFF_SCAN  = OFF_LEAF + (size_t)NBATCH*CELLS*4;    // int[4*65536] inclusive scan
static constexpr size_t OFF_BLKT  = OFF_SCAN + (size_t)NBATCH*CELLS*4;    // int[4*256] block totals
static constexpr size_t OFF_BLKO  = OFF_BLKT + 4096;                      // int[4*256] block exclusive offsets
static constexpr size_t OFF_CNT   = OFF_BLKO + 4096;                      // int[4] patches per image
static constexpr size_t OFF_EMB   = OFF_CNT  + 16;                        // int[4] emb float offsets
static constexpr size_t OFF_DST   = OFF_EMB  + 16;                        // int[4] dst float offsets
static constexpr size_t OFF_SZC   = OFF_DST  + 16;                        // int[4*6] per-(img,size) counts
static constexpr size_t OFF_LISTS = ((OFF_SZC + 96 + 255)/256)*256;       // int4[4*LISTPI] {r,c,dfs,0}

__global__ void k_init(int* szc){
    if (threadIdx.x < NBATCH*6) szc[threadIdx.x] = 0;
}

__global__ void k_cellsum(const float* __restrict__ edge, float* __restrict__ pyr){
    int t = blockIdx.x*blockDim.x + threadIdx.x;    // NBATCH*CELLS threads
    int b = t >> 16, cell = t & 65535;
    int cy = cell >> 8, cx = cell & 255;
    const float* e = edge + (size_t)b*IMGSZ*IMGSZ + (size_t)(cy*4)*IMGSZ + cx*4;
    float s = 0.f;
    #pragma unroll
    for (int i = 0; i < 4; i++){
        float4 v = *(const float4*)(e + (size_t)i*IMGSZ);
        s += v.x + v.y + v.z + v.w;
    }
    pyr[(size_t)b*PYRSTR + cell] = s;
}

__global__ void k_reduce(float* __restrict__ pyr, int srcOff, int dstOff, int dstN){
    int total = NBATCH*dstN*dstN;
    int t = blockIdx.x*blockDim.x + threadIdx.x;
    if (t >= total) return;
    int b = t/(dstN*dstN); int r = t - b*dstN*dstN;
    int y = r/dstN, x = r - y*dstN;
    int srcN = dstN*2;
    const float* s = pyr + (size_t)b*PYRSTR + srcOff;
    pyr[(size_t)b*PYRSTR + dstOff + y*dstN + x] =
        s[(2*y)*srcN + 2*x] + s[(2*y)*srcN + 2*x + 1] +
        s[(2*y+1)*srcN + 2*x] + s[(2*y+1)*srcN + 2*x + 1];
}

__device__ inline int morton8(int cy, int cx){
    int m = 0;
    #pragma unroll
    for (int i = 0; i < 8; i++)
        m |= (((cx>>i)&1) << (2*i)) | (((cy>>i)&1) << (2*i+1));
    return m;
}

__global__ void k_classify(const float* __restrict__ pyr, const int* __restrict__ thresP,
                           const int* __restrict__ minP, int* __restrict__ leaf){
    int t = blockIdx.x*blockDim.x + threadIdx.x;
    int b = t >> 16, cell = t & 65535;
    int cy = cell >> 8, cx = cell & 255;
    int thres = thresP[0];
    int mp = minP[0]; if (mp < 4) mp = 4;
    int twoMin = 2*mp;
    const float* P = pyr + (size_t)b*PYRSTR;
    int k = 8;
    while (k > 0){
        int s = 4 << k;
        if (s < twoMin) break;
        int n = 256 >> k;
        float sm = P[g_lvloff[k] + (cy>>k)*n + (cx>>k)];
        if (!((int)sm > thres)) break;   // exact: int(e.sum()) > thres
        k--;
    }
    int sz = 4 << k;
    int mask = (1 << k) - 1;
    int out = (((cy & mask) == 0) && ((cx & mask) == 0)) ? ((sz<<16)|(cy<<8)|cx) : 0;
    leaf[b*CELLS + morton8(cy, cx)] = out;
}

__global__ void k_scan1(const int* __restrict__ leaf, int* __restrict__ incl, int* __restrict__ blkTot){
    __shared__ int sh[256];
    int blk = blockIdx.x;            // 1024 blocks: b = blk>>8, seg = blk&255
    int b = blk >> 8, seg = blk & 255;
    int i = seg*256 + threadIdx.x;
    int f = (leaf[b*CELLS + i] != 0) ? 1 : 0;
    sh[threadIdx.x] = f; __syncthreads();
    for (int off = 1; off < 256; off <<= 1){
        int v = (threadIdx.x >= off) ? sh[threadIdx.x - off] : 0;
        __syncthreads();
        sh[threadIdx.x] += v;
        __syncthreads();
    }
    incl[b*CELLS + i] = sh[threadIdx.x];
    if (threadIdx.x == 255) blkTot[blk] = sh[255];
}

__global__ void k_scan2(const int* __restrict__ blkTot, int* __restrict__ blkOff, int* __restrict__ cnt){
    __shared__ int sh[256];
    int b = blockIdx.x;
    int v = blkTot[b*256 + threadIdx.x];
    sh[threadIdx.x] = v; __syncthreads();
    for (int off = 1; off < 256; off <<= 1){
        int u = (threadIdx.x >= off) ? sh[threadIdx.x - off] : 0;
        __syncthreads();
        sh[threadIdx.x] += u;
        __syncthreads();
    }
    blkOff[b*256 + threadIdx.x] = sh[threadIdx.x] - v;   // exclusive
    if (threadIdx.x == 255) cnt[b] = sh[255];
}

__global__ void k_final(const int* __restrict__ cnt, int* __restrict__ embOff, int* __restrict__ dstOff){
    int acc = 0, pre[NBATCH];
    for (int b = 0; b < NBATCH; b++){ pre[b] = acc; acc += cnt[b]; }
    for (int b = 0; b < NBATCH; b++){
        embOff[b] = pre[b]*DMODEL;
        dstOff[b] = acc*DMODEL + pre[b]*5;
    }
}

__global__ void k_compact(const int* __restrict__ leaf, const int* __restrict__ incl,
                          const int* __restrict__ blkOff, const int* __restrict__ embOff,
                          const int* __restrict__ dstOff, int* __restrict__ szc,
                          int* __restrict__ lists, float* __restrict__ out){
    int t = blockIdx.x*256 + threadIdx.x;
    int b = t >> 16, i = t & 65535;
    int packed = leaf[b*CELLS + i];
    if (!packed) return;
    int sz = packed >> 16, cy = (packed>>8)&255, cx = packed & 255;
    int dfs = incl[b*CELLS + i] - 1 + blkOff[b*256 + (i>>8)];
    int r = cy*4, c = cx*4;
    float* dd = out + dstOff[b] + (size_t)dfs*5;
    dd[0] = (float)IMGSZ; dd[1] = (float)IMGSZ; dd[2] = (float)sz; dd[3] = (float)r; dd[4] = (float)c;
    if (sz <= 128){
        int si = (31 - __clz(sz)) - 2;              // 4->0 ... 128->5
        int slot = atomicAdd(&szc[b*6 + si], 1);
        int* e = lists + ((size_t)b*LISTPI + g_capoff[si] + slot)*4;
        e[0] = r; e[1] = c; e[2] = dfs; e[3] = 0;
    } else {
        // size not in the weight table -> reference leaves zeros
        float* o = out + embOff[b] + (size_t)dfs*DMODEL;
        for (int d = 0; d < DMODEL; d++) o[d] = 0.f;
    }
}

constexpr int mtmax_c(int S){ return ((IMGSZ/S)*(IMGSZ/S) + 15)/16; }

// GEMM: tokens[m, d] = bias[d] + sum_k patch[m,k] * W[d,k], K = 3*S*S
// block = 128 threads (4 waves), tile M=16 patches, N=64 channels, K chunk 64.
// B (weight) tile is streamed global->LDS by the Tensor Data Mover; A tile is
// gathered by the threads in parallel with the DMA.
template<int S>
__global__ __launch_bounds__(128)
void k_gemm(const float* __restrict__ img, const float* __restrict__ W,
            const float* __restrict__ bias, const int* __restrict__ szc,
            const int* __restrict__ lists, const int* __restrict__ embOff,
            float* __restrict__ out){
    constexpr int K  = 3*S*S;
    constexpr int KC = 64;
    constexpr int LDK = KC + 4;   // stride 68 floats == 4 mod 64 -> conflict-free b64 frag reads
    constexpr int SI = (S==4)?0:(S==8)?1:(S==16)?2:(S==32)?3:(S==64)?4:5;
    constexpr int MT = mtmax_c(S);

    __shared__ float At[16][LDK];
    __shared__ float Bt[64][LDK];
    __shared__ int mr[16], mc[16], mdfs[16];

    int b  = blockIdx.x / MT;
    int mt = blockIdx.x - b*MT;
    int cnt = szc[b*6 + SI];
    if (mt*16 >= cnt) return;                 // uniform exit
    int ntile = blockIdx.y;
    int m0 = mt*16;

    int t = threadIdx.x;
    if (t < 16){
        int gm = m0 + t;
        if (gm < cnt){
            const int* e = lists + ((size_t)b*LISTPI + g_capoff[SI] + gm)*4;
            mr[t] = e[0]; mc[t] = e[1]; mdfs[t] = e[2];
        } else { mr[t] = 0; mc[t] = 0; mdfs[t] = -1; }
    }
    __syncthreads();

    int lane = t & 31, wave = t >> 5;
    int half = lane >> 4, l = lane & 15;
    int dBase = wave*16;

    v8f acc = {0,0,0,0,0,0,0,0};
    const float* imb = img + (size_t)b*3*IMGSZ*IMGSZ;
    const float* Wt  = W + (size_t)(ntile*64)*K;

#if USE_TDM
    // LDS byte offset of Bt (generic LDS pointer: low 32 bits are the LDS offset)
    uint32_t btLds = (uint32_t)(uintptr_t)(void*)&Bt[0][0];
    // D# group1 invariants: data_size=4B(2), pad_enable, pad every 64 DW by 4 DW,
    // tensor_dim1 = tile_dim1 = 64 rows, tile_dim0 = 64 elems, dim0 stride = K.
    const uint32_t g1dw0 = (2u<<16) | (1u<<20) | (5u<<22) | (3u<<25);
#endif

    for (int kc = 0; kc < K; kc += KC){
#if USE_TDM
        if (wave == 0){
            uint64_t ga = (uint64_t)(uintptr_t)(Wt + kc);
            uint32_t rem = (uint32_t)(K - kc);     // elements left in dim0 (OOB reads -> 0)
            v4u g0 = { 1u,                                       // count=1, user mode
                       btLds,                                    // lds_addr
                       (uint32_t)ga,                             // global_addr[31:0]
                       (uint32_t)((ga >> 32) & 0x01FFFFFFu) | 0x80000000u }; // addr[56:32] | type=2
            v8i g1 = { (int)g1dw0,
                       (int)((rem & 0xFFFFu) << 16),             // tensor_dim0[15:0]
                       (int)((rem >> 16) | (64u << 16)),         // tensor_dim0[31:16] | tensor_dim1=64
                       (int)(64u << 16),                         // tile_dim0 = 64
                       (int)64,                                  // tile_dim1 = 64
                       (int)K,                                   // tensor_dim0_stride = K
                       0, 0 };
            v4i gz = {0,0,0,0};
#if defined(__clang_major__) && (__clang_major__ >= 23)
            v8i gz8 = {0,0,0,0,0,0,0,0};
            __builtin_amdgcn_tensor_load_to_lds(g0, g1, gz, gz, gz8, 0);
#else
            __builtin_amdgcn_tensor_load_to_lds(g0, g1, gz, gz, 0);
#endif
        }
#endif
        {   // A tile: 16 patches x KC gathered from the image (overlaps the TDM)
            int m = t >> 3;
            int kk0 = (t & 7)*8;
            int r = mr[m], c = mc[m];
            #pragma unroll
            for (int q = 0; q < 8; q++){
                int kk = kk0 + q, k = kc + kk;
                float v = 0.f;
                if (k < K){
                    int ch = k/(S*S); int rem2 = k - ch*(S*S);
                    int i = rem2/S;   int j   = rem2 - i*S;
                    v = imb[(size_t)ch*IMGSZ*IMGSZ + (size_t)(r+i)*IMGSZ + (c+j)];
                }
                At[m][kk] = v;
            }
            int kn = kc + KC + kk0;
            if (kn < K){                        // prefetch next A chunk stream
                int ch = kn/(S*S); int rem2 = kn - ch*(S*S);
                int i = rem2/S;    int j   = rem2 - i*S;
                __builtin_prefetch(imb + (size_t)ch*IMGSZ*IMGSZ + (size_t)(r+i)*IMGSZ + (c+j), 0, 0);
            }
        }
#if USE_TDM
        if (wave == 0) __builtin_amdgcn_s_wait_tensorcnt(0);
#else
        {   // B tile: 64 channels x KC, contiguous rows of W (coalesced float4)
            int d = t >> 1;
            int kk0 = (t & 1)*32;
            const float* wr = Wt + (size_t)d*K + kc;
            #pragma unroll
            for (int q = 0; q < 32; q += 4){
                int kk = kk0 + q, k = kc + kk;
                float4 v = {0,0,0,0};
                if (k < K) v = *(const float4*)(wr + kk);
                *(float4*)&Bt[d][kk] = v;
            }
            __builtin_prefetch(wr + KC, 0, 0);
        }
#endif
        __syncthreads();
        #pragma unroll
        for (int kk = 0; kk < KC; kk += 4){
#if __has_builtin(__builtin_amdgcn_wmma_f32_16x16x4_f32)
            int ks = kk + 2*half;                // f32 16x16x4 fragment layout
            v2f a  = *(const v2f*)&At[l][ks];
            v2f bb = *(const v2f*)&Bt[dBase + l][ks];
            acc = __builtin_amdgcn_wmma_f32_16x16x4_f32(
                false, a, false, bb, (short)0, acc, false, false);
#else
            #pragma unroll
            for (int v8 = 0; v8 < 8; v8++){
                int m = v8 + 8*half;
                float s = acc[v8];
                for (int kq = 0; kq < 4; kq++) s += At[m][kk+kq]*Bt[dBase+l][kk+kq];
                acc[v8] = s;
            }
#endif
        }
        __syncthreads();
    }

    int col = ntile*64 + dBase + l;
    float bv = bias[col];
    int eo = embOff[b];
    #pragma unroll
    for (int v8 = 0; v8 < 8; v8++){
        int m = v8 + 8*half;                     // C/D layout: M = vgpr + 8*half, N = lane&15
        int dfs = mdfs[m];
        if (dfs >= 0)
            out[(size_t)eo + (size_t)dfs*DMODEL + col] = acc[v8] + bv;
    }
}

extern "C" void kernel_launch(void* const* d_in, const int* in_sizes, int n_in,
                              void* d_out, int out_size, void* d_ws, size_t ws_size,
                              hipStream_t stream){
    (void)in_sizes; (void)n_in; (void)out_size; (void)ws_size;
    const float* im   = (const float*)d_in[0];
    const float* edge = (const float*)d_in[1];
    const float* W[6]; const float* bs[6];
    for (int i = 0; i < 6; i++){ W[i] = (const float*)d_in[2 + 2*i]; bs[i] = (const float*)d_in[3 + 2*i]; }
    const int* thres = (const int*)d_in[14];
    const int* minp  = (const int*)d_in[15];

    char* ws = (char*)d_ws;
    float* pyr  = (float*)(ws + OFF_PYR);
    int* leaf   = (int*)(ws + OFF_LEAF);
    int* incl   = (int*)(ws + OFF_SCAN);
    int* blkT   = (int*)(ws + OFF_BLKT);
    int* blkO   = (int*)(ws + OFF_BLKO);
    int* cnt    = (int*)(ws + OFF_CNT);
    int* embO   = (int*)(ws + OFF_EMB);
    int* dstO   = (int*)(ws + OFF_DST);
    int* szc    = (int*)(ws + OFF_SZC);
    int* lists  = (int*)(ws + OFF_LISTS);
    float* out  = (float*)d_out;

    k_init<<<1, 64, 0, stream>>>(szc);
    k_cellsum<<<(NBATCH*CELLS)/256, 256, 0, stream>>>(edge, pyr);
    int dstN = 128;
    for (int k = 1; k <= 8; k++){
        int total = NBATCH*dstN*dstN;
        k_reduce<<<(total + 255)/256, 256, 0, stream>>>(pyr, h_lvloff[k-1], h_lvloff[k], dstN);
        dstN >>= 1;
    }
    k_classify<<<(NBATCH*CELLS)/256, 256, 0, stream>>>(pyr, thres, minp, leaf);
    k_scan1<<<NBATCH*256, 256, 0, stream>>>(leaf, incl, blkT);
    k_scan2<<<NBATCH, 256, 0, stream>>>(blkT, blkO, cnt);
    k_final<<<1, 1, 0, stream>>>(cnt, embO, dstO);
    k_compact<<<NBATCH*256, 256, 0, stream>>>(leaf, incl, blkO, embO, dstO, szc, lists, out);

    dim3 blk(128);
    { dim3 g(NBATCH*mtmax_c(4),   DMODEL/64); k_gemm<4>  <<<g, blk, 0, stream>>>(im, W[0], bs[0], szc, lists, embO, out); }
    { dim3 g(NBATCH*mtmax_c(8),   DMODEL/64); k_gemm<8>  <<<g, blk, 0, stream>>>(im, W[1], bs[1], szc, lists, embO, out); }
    { dim3 g(NBATCH*mtmax_c(16),  DMODEL/64); k_gemm<16> <<<g, blk, 0, stream>>>(im, W[2], bs[2], szc, lists, embO, out); }
    { dim3 g(NBATCH*mtmax_c(32),  DMODEL/64); k_gemm<32> <<<g, blk, 0, stream>>>(im, W[3], bs[3], szc, lists, embO, out); }
    { dim3 g(NBATCH*mtmax_c(64),  DMODEL/64); k_gemm<64> <<<g, blk, 0, stream>>>(im, W[4], bs[4], szc, lists, embO, out); }
    { dim3 g(NBATCH*mtmax_c(128), DMODEL/64); k_gemm<128><<<g, blk, 0, stream>>>(im, W[5], bs[5], szc, lists, embO, out); }
}